// DYSPN_25074019074066
// MI455X (gfx1250) — compile-verified
//
#include <hip/hip_runtime.h>
#include <hip/hip_bf16.h>

// ---------------------------------------------------------------------------
// DYSPN propagation, MI455X (gfx1250).
//
// Bandwidth-bound: ~235 MB moved / 23.3 TB/s ~= 10 us floor. FLOPs ~0.3 G
// (so WMMA is inapplicable; the win is data movement).
// Strategy:
//   * fold/unfold collapsed analytically: out_fold(p) = cs(p) * sum_k
//     kernel(k, p + (3-i, 3-j))   (OOB shifted terms dropped).
//   * INDEX[k] == 3 - max(|3-i|,|3-j|)  (Chebyshev rings), center masked.
//   * affinity streamed with coalesced b128 loads (each element read 2x,
//     +-3-row reuse window -> resident in 192MB L2, HBM sees it once).
//     k-loop split at plane 25 so every access stays within the +-8MB
//     24-bit instruction offset (SGPR base + voffset + imm form).
//   * attention halo tiles staged to LDS with CDNA5 async global->LDS copies
//     (ASYNCcnt / s_wait_asynccnt), freeing the VMEM path for affinity.
//   * single-use streams (cs/co/out) use non-temporal hints to avoid
//     polluting the L2 affinity reuse window.
// ---------------------------------------------------------------------------

typedef float v4f    __attribute__((ext_vector_type(4)));
typedef float v4f_a4 __attribute__((ext_vector_type(4), aligned(4)));

#define HH 256
#define WW 256
#define HWN (HH * WW)
#define TW 128
#define TH 8
#define AROWS (TH + 6)     // 14
#define ACOLS (TW + 6)     // 134
#define APITCH (ACOLS + 2) // 136 (bank-spread pad)

typedef __attribute__((address_space(1))) int* gint_ptr;
typedef __attribute__((address_space(3))) int* lint_ptr;

// ---- CDNA5 async global->LDS copy (ASYNCcnt path) -------------------------
__device__ __forceinline__ void async_g2l_b32(const float* gsrc, float* ldst) {
#if __has_builtin(__builtin_amdgcn_global_load_async_to_lds_b32)
  __builtin_amdgcn_global_load_async_to_lds_b32(
      (gint_ptr)(int*)gsrc, (lint_ptr)(int*)ldst, 0, 0);
#else
  unsigned loff = (unsigned)(unsigned long long)((lint_ptr)(int*)ldst);
  asm volatile("global_load_async_to_lds_b32 %0, %1, off"
               :
               : "v"(loff), "v"(gsrc)
               : "memory");
#endif
}

__device__ __forceinline__ void wait_async0() {
#if __has_builtin(__builtin_amdgcn_s_wait_asynccnt)
  __builtin_amdgcn_s_wait_asynccnt(0);
#else
  asm volatile("s_wait_asynccnt 0" ::: "memory");
#endif
}

// ---------------------------------------------------------------------------
__device__ __forceinline__ v4f vabs4(v4f a) {
  v4f r;
  r.x = __builtin_fabsf(a.x);
  r.y = __builtin_fabsf(a.y);
  r.z = __builtin_fabsf(a.z);
  r.w = __builtin_fabsf(a.w);
  return r;
}

// affinity plane read at (y, x..x+3), zero outside the image.
__device__ __forceinline__ v4f loadAff4(const float* __restrict__ plane,
                                        int y, int x) {
  v4f r = {0.f, 0.f, 0.f, 0.f};
  if ((unsigned)y < (unsigned)HH) {
    const float* row = plane + y * WW;
    if ((unsigned)x <= (unsigned)(WW - 4)) {
      r = *(const v4f_a4*)(row + x);  // dword-aligned b128 is legal on gfx
    } else {
      if ((unsigned)(x + 0) < (unsigned)WW) r.x = row[x + 0];
      if ((unsigned)(x + 1) < (unsigned)WW) r.y = row[x + 1];
      if ((unsigned)(x + 2) < (unsigned)WW) r.z = row[x + 2];
      if ((unsigned)(x + 3) < (unsigned)WW) r.w = row[x + 3];
    }
  }
  return r;
}

__device__ __forceinline__ v4f ldsAtt4(const float* s) {
  v4f r;
  r.x = s[0]; r.y = s[1]; r.z = s[2]; r.w = s[3];
  return r;
}

__global__ __launch_bounds__(256) void dyspn_kernel(
    const float* __restrict__ aff, const float* __restrict__ att,
    const float* __restrict__ cs, const float* __restrict__ co,
    float* __restrict__ out) {
  __shared__ float sAtt[4 * AROWS * APITCH];

  const int tile_x = blockIdx.x * TW;
  const int tile_y = blockIdx.y * TH;
  const int b = blockIdx.z;
  const int tid = threadIdx.x;

  const float* attB = att + (size_t)b * 4 * HWN;
  const float* affB = aff + (size_t)b * 49 * HWN;
  const float* affM = affB + (size_t)25 * HWN;  // rebased: keeps imm < 8MB

  // ---- stage attention (4 planes, +-3 halo) into LDS via async copies ----
  for (int i = tid; i < 4 * AROWS * APITCH; i += 256) sAtt[i] = 0.0f;
  __syncthreads();

  const int TOT = 4 * AROWS * ACOLS;
  for (int idx = tid; idx < TOT; idx += 256) {
    int pl = idx / (AROWS * ACOLS);
    int rem = idx - pl * (AROWS * ACOLS);
    int ly = rem / ACOLS;
    int lx = rem - ly * ACOLS;
    int gy = tile_y - 3 + ly;
    int gx = tile_x - 3 + lx;
    if ((unsigned)gy < (unsigned)HH && (unsigned)gx < (unsigned)WW) {
      async_g2l_b32(attB + pl * HWN + gy * WW + gx,
                    &sAtt[(pl * AROWS + ly) * APITCH + lx]);
    }
  }
  wait_async0();
  __syncthreads();

  // ---- per-thread: 4 consecutive pixels on one row -----------------------
  const int lx4 = (tid & 31) * 4;  // 0..124
  const int lyy = tid >> 5;        // 0..7
  const int x0 = tile_x + lx4;
  const int y = tile_y + lyy;

  // attention at p for rings 0..2 and self (plane 3), from LDS
  const v4f w0 = ldsAtt4(&sAtt[(0 * AROWS + lyy + 3) * APITCH + lx4 + 3]);
  const v4f w1 = ldsAtt4(&sAtt[(1 * AROWS + lyy + 3) * APITCH + lx4 + 3]);
  const v4f w2 = ldsAtt4(&sAtt[(2 * AROWS + lyy + 3) * APITCH + lx4 + 3]);
  const v4f w3 = ldsAtt4(&sAtt[(3 * AROWS + lyy + 3) * APITCH + lx4 + 3]);

  v4f S = {0.f, 0.f, 0.f, 0.f};   // sum gath * affinity
  v4f Sp = {0.f, 0.f, 0.f, 0.f};  // sum gath * |affinity|
  v4f K = {0.f, 0.f, 0.f, 0.f};   // sum_k kernel(k, p + delta_k)

#pragma unroll
  for (int k = 0; k < 49; ++k) {
    if (k == 24) continue;  // center masked (INDEX==3)
    const int i = k / 7, j = k % 7;
    const int dy = 3 - i, dx = 3 - j;
    const int ady = dy < 0 ? -dy : dy;
    const int adx = dx < 0 ? -dx : dx;
    const int r = 3 - (ady > adx ? ady : adx);  // ring 0..2
    // rebased plane pointer so all byte offsets fit the 24-bit imm
    const float* plane =
        (k < 25) ? (affB + k * HWN) : (affM + (k - 25) * HWN);

    if ((k & 7) == 0 && k + 16 < 49)  // gfx1250 global_prefetch for stream
      __builtin_prefetch(plane + 16 * HWN + y * WW + x0, 0, 0);

    // local use: S_ppt / S_prime contributions at p (always in-image)
    const v4f a = *(const v4f*)(plane + y * WW + x0);
    const v4f wl = (r == 0) ? w0 : ((r == 1) ? w1 : w2);
    S += a * wl;
    Sp += vabs4(a) * wl;

    // shifted use: kernel(k, p + delta), OOB -> 0 (fold semantics)
    const v4f ash = loadAff4(plane, y + dy, x0 + dx);
    const v4f wsh =
        ldsAtt4(&sAtt[(r * AROWS + lyy + 3 + dy) * APITCH + lx4 + 3 + dx]);
    K += ash * wsh;
  }

  S += w3;   // S_ppt = conv + att_self
  Sp += w3;  // S_prime = conv_abs + att_self
  const v4f den = Sp + 1e-6f;

  const size_t pbase = (size_t)b * HWN + (size_t)y * WW + x0;
  // single-use streams: non-temporal to keep L2 for affinity reuse
  const v4f cs4 = __builtin_nontemporal_load((const v4f*)(cs + pbase));
  const v4f co4 = __builtin_nontemporal_load((const v4f*)(co + pbase));

  const v4f num = (K + w3) * cs4 + (den - S) * co4;
  const v4f o = num / den;
  __builtin_nontemporal_store(o, (v4f*)(out + pbase));
}

// ---------------------------------------------------------------------------
extern "C" void kernel_launch(void* const* d_in, const int* in_sizes, int n_in,
                              void* d_out, int out_size, void* d_ws,
                              size_t ws_size, hipStream_t stream) {
  const float* aff = (const float*)d_in[0];  // (B,49,256,256) f32
  const float* att = (const float*)d_in[1];  // (B, 4,256,256) f32
  const float* cs = (const float*)d_in[2];   // (B, 1,256,256) f32
  const float* co = (const float*)d_in[3];   // (B, 1,256,256) f32
  float* out = (float*)d_out;                // (B, 1,256,256) f32

  const int B = in_sizes[2] / (HH * WW);
  dim3 grid(WW / TW, HH / TH, B);  // (2, 32, B)
  dyspn_kernel<<<grid, dim3(256, 1, 1), 0, stream>>>(aff, att, cs, co, out);
}